// LSTMTagger_44272522887513
// MI455X (gfx1250) — compile-verified
//
#include <hip/hip_runtime.h>
#include <math.h>
#include <stdint.h>

// ---------------------------------------------------------------------------
// LSTM tagger for MI455X (gfx1250, wave32).
//   S=2048, LC=12, E=256, H=256, EC=64, HC=128, V=50000, VC=128, T=50
//
// Stages:
//   A) WMMA GEMM: Gc[24576,512]  = char_emb[word_chars] @ Wih_c^T + b     (parallel)
//   B) persistent 1-WG char LSTM recurrence (weights in VGPRs, h in LDS,
//      next-step input row staged via GLOBAL_LOAD_ASYNC_TO_LDS ping-pong) (serial)
//   C) WMMA GEMM: Gw[2048,1024]  = word_emb[sentence] @ Wih_w[:,:256]^T+b (parallel)
//   D) WMMA GEMM: Gw            += char_hid @ Wih_w[:,256:]^T             (parallel)
//   E) persistent 4-WG word LSTM recurrence (1MB Whh_w split into VGPRs,
//      async-staged Gw slices, 4KB gate exchange through L2 + grid barrier)(serial)
//   F) tag projection + log_softmax                                       (parallel)
// ---------------------------------------------------------------------------

typedef __attribute__((ext_vector_type(2))) float v2f;
typedef __attribute__((ext_vector_type(8))) float v8f;

#define S_LEN 2048
#define LC_N  12
#define EC_N  64
#define HC_N  128
#define E_N   256
#define H_N   256
#define T_N   50
#define GC_W  (4 * HC_N)   /* 512  */
#define GW_W  (4 * H_N)    /* 1024 */

__device__ __forceinline__ float sigf(float x) {
  return 1.0f / (1.0f + __expf(-x));
}

// ---- CDNA5 async global->LDS staging (guarded; falls back to sync loads) --
#if __has_builtin(__builtin_amdgcn_global_load_async_to_lds_b32)
#define ASYNC_LDS 1
typedef __attribute__((address_space(1))) int* glb_i32p;
typedef __attribute__((address_space(3))) int* lds_i32p;
__device__ __forceinline__ void async_ld_f32(float* lds, const float* g) {
  __builtin_amdgcn_global_load_async_to_lds_b32(
      (glb_i32p)(uintptr_t)g, (lds_i32p)(unsigned)(uintptr_t)lds, 0, 0);
}
__device__ __forceinline__ void wait_async_le1() {
#if __has_builtin(__builtin_amdgcn_s_wait_asynccnt)
  __builtin_amdgcn_s_wait_asynccnt(1);
#else
  asm volatile("s_wait_asynccnt 0x1" ::: "memory");
#endif
}
#else
#define ASYNC_LDS 0
#endif

// ---------------------------------------------------------------------------
// Stage A: Gc = gather(char_emb, word_chars) @ Wih_c^T + (bih_c + bhh_c)
// M=24576, N=512, K=64.  One wave per 16x16 tile, V_WMMA_F32_16X16X4_F32.
// ---------------------------------------------------------------------------
__global__ __launch_bounds__(256) void gemm_char_in(
    const int* __restrict__ word_chars, const float* __restrict__ char_emb,
    const float* __restrict__ Wih_c, const float* __restrict__ bih_c,
    const float* __restrict__ bhh_c, float* __restrict__ Gc) {
  const int lane = threadIdx.x & 31;
  const int wave = threadIdx.x >> 5;
  const int tile = blockIdx.x * 8 + wave;          // 49152 tiles
  const int tm = tile >> 5;                        // 0..1535
  const int tn = tile & 31;                        // 0..31
  const int half = lane >> 4;
  const int l16  = lane & 15;

  const int row = tm * 16 + l16;                   // A row (0..24575)
  const int ci  = word_chars[row];
  const float* __restrict__ arow = char_emb + (size_t)ci * EC_N;
  const int col = tn * 16 + l16;                   // B column = gate index
  const float* __restrict__ bcol = Wih_c + (size_t)col * EC_N;
  const int ks = half * 2;

  v8f c = {};
#pragma unroll
  for (int k0 = 0; k0 < EC_N; k0 += 4) {
    v2f a, b;
    a.x = arow[k0 + ks];  a.y = arow[k0 + ks + 1];
    b.x = bcol[k0 + ks];  b.y = bcol[k0 + ks + 1];
    c = __builtin_amdgcn_wmma_f32_16x16x4_f32(false, a, false, b,
                                              (short)0, c, false, false);
  }
  const float bias = bih_c[col] + bhh_c[col];
  const int rbase = tm * 16 + half * 8;
#pragma unroll
  for (int v = 0; v < 8; ++v)
    Gc[(size_t)(rbase + v) * GC_W + col] = c[v] + bias;
}

// ---------------------------------------------------------------------------
// Stage C: Gw = word_emb[sentence] @ Wih_w[:, :256]^T + (bih_w + bhh_w)
// M=2048, N=1024, K=256.
// ---------------------------------------------------------------------------
__global__ __launch_bounds__(256) void gemm_word_in(
    const int* __restrict__ sentence, const float* __restrict__ word_emb,
    const float* __restrict__ Wih_w, const float* __restrict__ bih_w,
    const float* __restrict__ bhh_w, float* __restrict__ Gw) {
  const int lane = threadIdx.x & 31;
  const int wave = threadIdx.x >> 5;
  const int tile = blockIdx.x * 8 + wave;          // 8192 tiles
  const int tm = tile >> 6;                        // 0..127
  const int tn = tile & 63;                        // 0..63
  const int half = lane >> 4;
  const int l16  = lane & 15;

  const int row = tm * 16 + l16;                   // sentence position
  const int si  = sentence[row];
  const float* __restrict__ arow = word_emb + (size_t)si * E_N;
  const int col = tn * 16 + l16;
  const float* __restrict__ bcol = Wih_w + (size_t)col * (E_N + HC_N);
  const int ks = half * 2;

  v8f c = {};
#pragma unroll
  for (int k0 = 0; k0 < E_N; k0 += 4) {
    v2f a, b;
    a.x = arow[k0 + ks];  a.y = arow[k0 + ks + 1];
    b.x = bcol[k0 + ks];  b.y = bcol[k0 + ks + 1];
    c = __builtin_amdgcn_wmma_f32_16x16x4_f32(false, a, false, b,
                                              (short)0, c, false, false);
  }
  const float bias = bih_w[col] + bhh_w[col];
  const int rbase = tm * 16 + half * 8;
#pragma unroll
  for (int v = 0; v < 8; ++v)
    Gw[(size_t)(rbase + v) * GW_W + col] = c[v] + bias;
}

// ---------------------------------------------------------------------------
// Stage D: Gw += char_hid @ Wih_w[:, 256:384]^T   (accumulating GEMM)
// M=2048, N=1024, K=128.
// ---------------------------------------------------------------------------
__global__ __launch_bounds__(256) void gemm_word_in2(
    const float* __restrict__ CH, const float* __restrict__ Wih_w,
    float* __restrict__ Gw) {
  const int lane = threadIdx.x & 31;
  const int wave = threadIdx.x >> 5;
  const int tile = blockIdx.x * 8 + wave;
  const int tm = tile >> 6;
  const int tn = tile & 63;
  const int half = lane >> 4;
  const int l16  = lane & 15;

  const int row = tm * 16 + l16;
  const float* __restrict__ arow = CH + (size_t)row * HC_N;
  const int col = tn * 16 + l16;
  const float* __restrict__ bcol = Wih_w + (size_t)col * (E_N + HC_N) + E_N;
  const int ks = half * 2;
  const int rbase = tm * 16 + half * 8;

  v8f c;
#pragma unroll
  for (int v = 0; v < 8; ++v)
    c[v] = Gw[(size_t)(rbase + v) * GW_W + col];

#pragma unroll
  for (int k0 = 0; k0 < HC_N; k0 += 4) {
    v2f a, b;
    a.x = arow[k0 + ks];  a.y = arow[k0 + ks + 1];
    b.x = bcol[k0 + ks];  b.y = bcol[k0 + ks + 1];
    c = __builtin_amdgcn_wmma_f32_16x16x4_f32(false, a, false, b,
                                              (short)0, c, false, false);
  }
#pragma unroll
  for (int v = 0; v < 8; ++v)
    Gw[(size_t)(rbase + v) * GW_W + col] = c[v];
}

// ---------------------------------------------------------------------------
// Stage B: char LSTM recurrence.  One persistent workgroup, 1024 threads.
// Thread (r,q): r = gate row (0..511), q = column half (0/1, 64 cols each).
// Whh_c row slice lives in VGPRs; h broadcast from LDS; masked steps skipped.
// Next step's 2KB Gc row is async-staged into an LDS ping-pong buffer by
// thread tid (producer == consumer per element), hidden behind the matvec.
// ---------------------------------------------------------------------------
__global__ __launch_bounds__(1024) void char_rnn(
    const float* __restrict__ Gc, const float* __restrict__ Whh_c,
    const int* __restrict__ word_lens, float* __restrict__ CH) {
  const int tid = threadIdx.x;
  const int r = tid >> 1;
  const int q = tid & 1;

  v2f w[32];
  const v2f* __restrict__ wp =
      (const v2f*)(Whh_c + (size_t)r * HC_N + q * 64);
#pragma unroll
  for (int k = 0; k < 32; ++k) w[k] = wp[k];

  __shared__ __align__(16) float hS[HC_N];
  __shared__ __align__(16) float cS[HC_N];
  __shared__ float part[1024];
  __shared__ float gates[GC_W];
#if ASYNC_LDS
  __shared__ __align__(16) float gbuf[2][GC_W];
#endif

  if (tid < HC_N) { hS[tid] = 0.0f; cS[tid] = 0.0f; }
  __syncthreads();

#if ASYNC_LDS
  int b = 0;
  if (tid < GC_W) async_ld_f32(&gbuf[0][tid], Gc + tid);   // row (t=0,lc=0)
#endif

  for (int t = 0; t < S_LEN; ++t) {
    const int len = word_lens[t];
    const float* __restrict__ gc = Gc + (size_t)t * LC_N * GC_W;
    for (int lc = 0; lc < len; ++lc) {
#if ASYNC_LDS
      // issue async fetch of the NEXT valid row (clamped at sequence end)
      if (tid < GC_W) {
        size_t nrow;
        if (lc + 1 < len)          nrow = (size_t)t * LC_N + lc + 1;
        else if (t + 1 < S_LEN)    nrow = (size_t)(t + 1) * LC_N;
        else                       nrow = (size_t)t * LC_N + lc;
        async_ld_f32(&gbuf[b ^ 1][tid], Gc + nrow * GC_W + tid);
      }
#else
      if (tid < 16 && lc + 1 < len)
        __builtin_prefetch(gc + (lc + 1) * GC_W + tid * 32, 0, 0);
#endif

      const v2f* __restrict__ hp = (const v2f*)(hS + q * 64);
      v2f acc2 = {0.0f, 0.0f};
#pragma unroll
      for (int k = 0; k < 32; ++k) acc2 = w[k] * hp[k] + acc2;
#if ASYNC_LDS
      part[tid] = acc2.x + acc2.y;
#else
      part[tid] = acc2.x + acc2.y + ((q == 0) ? gc[lc * GC_W + r] : 0.0f);
#endif
      __syncthreads();

      if (tid < GC_W) {
#if ASYNC_LDS
        wait_async_le1();   // wave-uniform (waves 0-15); covers own elements
        gates[tid] = part[2 * tid] + part[2 * tid + 1] + gbuf[b][tid];
#else
        gates[tid] = part[2 * tid] + part[2 * tid + 1];
#endif
      }
      __syncthreads();

      if (tid < HC_N) {
        const float iv = sigf(gates[tid]);
        const float fv = sigf(gates[HC_N + tid]);
        const float gv = tanhf(gates[2 * HC_N + tid]);
        const float ov = sigf(gates[3 * HC_N + tid]);
        const float cn = fv * cS[tid] + iv * gv;
        cS[tid] = cn;
        hS[tid] = ov * tanhf(cn);
      }
      __syncthreads();
#if ASYNC_LDS
      b ^= 1;
#endif
    }
    if (tid < HC_N) CH[(size_t)t * HC_N + tid] = hS[tid];
  }
}

// ---------------------------------------------------------------------------
// Stage E: word LSTM recurrence.  4 persistent workgroups (one gate group
// each, 256KB of Whh_w in VGPRs per WG).  Gate activations exchanged through
// L2 (double buffered) with a sense-reversing grid barrier; each WG keeps a
// redundant copy of (h, c) so only one global sync per step is needed.
// The WG's 1KB Gw slice for the next step is async-staged into LDS.
// ---------------------------------------------------------------------------
__global__ __launch_bounds__(1024) void word_rnn(
    const float* __restrict__ Gw, const float* __restrict__ Whh_w,
    float* __restrict__ gatebuf, unsigned* __restrict__ syncv,
    float* __restrict__ lstm_out) {
  const int tid = threadIdx.x;
  const int g = blockIdx.x;                 // gate group: 0=i 1=f 2=g 3=o
  const int r = tid >> 2;                   // 0..255 row within group
  const int q = tid & 3;                    // 0..3 column quarter (64 cols)
  const int j = g * H_N + r;                // global gate row

  v2f w[32];
  const v2f* __restrict__ wp =
      (const v2f*)(Whh_w + (size_t)j * H_N + q * 64);
#pragma unroll
  for (int k = 0; k < 32; ++k) w[k] = wp[k];

  __shared__ __align__(16) float hS[H_N];
  __shared__ float cS[H_N];
  __shared__ float part[1024];
  __shared__ unsigned lsense;
#if ASYNC_LDS
  __shared__ __align__(16) float gwb[2][H_N];
#endif

  if (tid < H_N) { hS[tid] = 0.0f; cS[tid] = 0.0f; }
  if (tid == 0) lsense = 0u;
  __syncthreads();

  unsigned* cnt   = syncv;
  unsigned* sense = syncv + 1;

#if ASYNC_LDS
  int b = 0;
  if (q == 0) async_ld_f32(&gwb[0][r], Gw + (size_t)g * H_N + r);  // row t=0
#endif

  for (int t = 0; t < S_LEN; ++t) {
    const int p = t & 1;
#if ASYNC_LDS
    if (q == 0) {
      const int nt = (t + 1 < S_LEN) ? t + 1 : t;
      async_ld_f32(&gwb[b ^ 1][r], Gw + (size_t)nt * GW_W + g * H_N + r);
    }
#endif

    const v2f* __restrict__ hp = (const v2f*)(hS + q * 64);
    v2f acc2 = {0.0f, 0.0f};
#pragma unroll
    for (int k = 0; k < 32; ++k) acc2 = w[k] * hp[k] + acc2;

#if ASYNC_LDS
    wait_async_le1();   // per-wave; covers this wave's own staged elements
    part[tid] = acc2.x + acc2.y + ((q == 0) ? gwb[b][r] : 0.0f);
#else
    part[tid] = acc2.x + acc2.y +
                ((q == 0) ? Gw[(size_t)t * GW_W + j] : 0.0f);
#endif
    __syncthreads();

    if (tid < H_N) {
      const float s = part[tid * 4] + part[tid * 4 + 1] +
                      part[tid * 4 + 2] + part[tid * 4 + 3];
      const float a = (g == 2) ? tanhf(s) : sigf(s);
      __hip_atomic_store(&gatebuf[p * GW_W + g * H_N + tid], a,
                         __ATOMIC_RELAXED, __HIP_MEMORY_SCOPE_AGENT);
    }

    // ---- grid barrier (sense reversing) ----
    __syncthreads();
    if (tid == 0) {
      __threadfence();
      const unsigned ls = lsense ^ 1u;
      lsense = ls;
      if (atomicAdd(cnt, 1u) == 3u) {
        __hip_atomic_store(cnt, 0u, __ATOMIC_RELAXED, __HIP_MEMORY_SCOPE_AGENT);
        __threadfence();
        __hip_atomic_store(sense, ls, __ATOMIC_RELEASE, __HIP_MEMORY_SCOPE_AGENT);
      } else {
        while (__hip_atomic_load(sense, __ATOMIC_ACQUIRE,
                                 __HIP_MEMORY_SCOPE_AGENT) != ls) {}
      }
    }
    __syncthreads();
    // ----------------------------------------

    if (tid < H_N) {
      const float* gb = gatebuf + p * GW_W;
      const float iv = __hip_atomic_load(&gb[tid], __ATOMIC_RELAXED,
                                         __HIP_MEMORY_SCOPE_AGENT);
      const float fv = __hip_atomic_load(&gb[H_N + tid], __ATOMIC_RELAXED,
                                         __HIP_MEMORY_SCOPE_AGENT);
      const float gv = __hip_atomic_load(&gb[2 * H_N + tid], __ATOMIC_RELAXED,
                                         __HIP_MEMORY_SCOPE_AGENT);
      const float ov = __hip_atomic_load(&gb[3 * H_N + tid], __ATOMIC_RELAXED,
                                         __HIP_MEMORY_SCOPE_AGENT);
      const float cn = fv * cS[tid] + iv * gv;  // activations pre-applied
      cS[tid] = cn;
      const float hn = ov * tanhf(cn);
      hS[tid] = hn;
      if (g == 0) lstm_out[(size_t)t * H_N + tid] = hn;
    }
    __syncthreads();
#if ASYNC_LDS
    b ^= 1;
#endif
  }
}

// ---------------------------------------------------------------------------
// Stage F: tag projection + log_softmax.  One 64-thread block per position.
// ---------------------------------------------------------------------------
__global__ __launch_bounds__(64) void tag_logsoftmax(
    const float* __restrict__ lstm_out, const float* __restrict__ W_tag,
    const float* __restrict__ b_tag, float* __restrict__ out) {
  const int t = blockIdx.x;
  const int j = threadIdx.x;
  __shared__ float red[64];
  __shared__ float xs[H_N];

  for (int k = j; k < H_N; k += 64) xs[k] = lstm_out[(size_t)t * H_N + k];
  __syncthreads();

  float dot = -1e30f;
  if (j < T_N) {
    float s = b_tag[j];
    const float* __restrict__ wr = W_tag + (size_t)j * H_N;
#pragma unroll 8
    for (int k = 0; k < H_N; ++k) s += wr[k] * xs[k];
    dot = s;
  }
  red[j] = dot;
  __syncthreads();
#pragma unroll
  for (int s = 32; s > 0; s >>= 1) {
    if (j < s) red[j] = fmaxf(red[j], red[j + s]);
    __syncthreads();
  }
  const float m = red[0];
  __syncthreads();
  red[j] = (j < T_N) ? __expf(dot - m) : 0.0f;
  __syncthreads();
#pragma unroll
  for (int s = 32; s > 0; s >>= 1) {
    if (j < s) red[j] += red[j + s];
    __syncthreads();
  }
  const float lse = m + __logf(red[0]);
  if (j < T_N) out[(size_t)t * T_N + j] = dot - lse;
}

// ---------------------------------------------------------------------------
extern "C" void kernel_launch(void* const* d_in, const int* in_sizes, int n_in,
                              void* d_out, int out_size, void* d_ws,
                              size_t ws_size, hipStream_t stream) {
  const int*   sentence   = (const int*)d_in[0];
  const int*   word_chars = (const int*)d_in[1];
  const int*   word_lens  = (const int*)d_in[2];
  const float* word_emb   = (const float*)d_in[5];
  const float* char_emb   = (const float*)d_in[6];
  const float* Wih_c      = (const float*)d_in[7];
  const float* Whh_c      = (const float*)d_in[8];
  const float* bih_c      = (const float*)d_in[9];
  const float* bhh_c      = (const float*)d_in[10];
  const float* Wih_w      = (const float*)d_in[11];
  const float* Whh_w      = (const float*)d_in[12];
  const float* bih_w      = (const float*)d_in[13];
  const float* bhh_w      = (const float*)d_in[14];
  const float* W_tag      = (const float*)d_in[15];
  const float* b_tag      = (const float*)d_in[16];
  float* out = (float*)d_out;

  char* ws = (char*)d_ws;
  const size_t oGc = 0;                                        // 24576*512 f32
  const size_t oGw = oGc + (size_t)S_LEN * LC_N * GC_W * 4;    // 2048*1024 f32
  const size_t oCH = oGw + (size_t)S_LEN * GW_W * 4;           // 2048*128 f32
  const size_t oLO = oCH + (size_t)S_LEN * HC_N * 4;           // 2048*256 f32
  const size_t oGB = oLO + (size_t)S_LEN * H_N * 4;            // 2*1024 f32
  const size_t oSY = oGB + 2 * (size_t)GW_W * 4;               // sync words

  float*    Gc      = (float*)(ws + oGc);
  float*    Gw      = (float*)(ws + oGw);
  float*    CH      = (float*)(ws + oCH);
  float*    LO      = (float*)(ws + oLO);
  float*    gatebuf = (float*)(ws + oGB);
  unsigned* syncv   = (unsigned*)(ws + oSY);

  // barrier state must be zero at every launch (graph-capture safe memset)
  (void)hipMemsetAsync(syncv, 0, 2 * sizeof(unsigned), stream);

  gemm_char_in<<<6144, 256, 0, stream>>>(word_chars, char_emb, Wih_c, bih_c,
                                         bhh_c, Gc);
  gemm_word_in<<<1024, 256, 0, stream>>>(sentence, word_emb, Wih_w, bih_w,
                                         bhh_w, Gw);
  char_rnn<<<1, 1024, 0, stream>>>(Gc, Whh_c, word_lens, CH);
  gemm_word_in2<<<1024, 256, 0, stream>>>(CH, Wih_w, Gw);
  word_rnn<<<4, 1024, 0, stream>>>(Gw, Whh_w, gatebuf, syncv, LO);
  tag_logsoftmax<<<2048, 64, 0, stream>>>(LO, W_tag, b_tag, out);
}